// KeySegPred_62165356642522
// MI455X (gfx1250) — compile-verified
//
#include <hip/hip_runtime.h>
#include <math.h>

typedef float v2f __attribute__((ext_vector_type(2)));
typedef float v8f __attribute__((ext_vector_type(8)));

#define D_SEG   128
#define D_S     258
#define D_M7    512
#define D_M8    256
#define MASK_N  64
#define T_SLOTS 96
#define BATCH   4096
#define DW_COLS 252   // D_M8 - CAND_OUT

static __device__ __forceinline__ v8f wmma_f32(v2f a, v2f b, v8f c) {
  // V_WMMA_F32_16X16X4_F32 : D(16x16 f32) = A(16x4 f32) * B(4x16 f32) + C
  return __builtin_amdgcn_wmma_f32_16x16x4_f32(
      /*neg_a=*/false, a, /*neg_b=*/false, b,
      /*c_mod=*/(short)0, c, /*reuse_a=*/false, /*reuse_b=*/false);
}

// ---------------------------------------------------------------------------
// Kernel 1: fused gather + GEMM1 (code @ w1^T + b1) + BatchNorm + ReLU
// One wave32 per 16x16 tile of h (4096 x 512).
// ---------------------------------------------------------------------------
__global__ __launch_bounds__(32) void k_gemm1(
    const int* __restrict__ src, const int* __restrict__ dest,
    const float* __restrict__ offset,
    const float* __restrict__ emb_feat, const float* __restrict__ emb_dest,
    const float* __restrict__ w1, const float* __restrict__ b1,
    const float* __restrict__ bn_gamma, const float* __restrict__ bn_beta,
    const float* __restrict__ bn_mean, const float* __restrict__ bn_var,
    float* __restrict__ hbuf)
{
  const int n0   = blockIdx.x * 16;     // output column tile (0..511)
  const int m0   = blockIdx.y * 16;     // batch row tile     (0..4095)
  const int lane = threadIdx.x & 31;
  const int half = lane >> 4;           // 0: K-pairs {0,1}; 1: K-pairs {2,3}
  const int l16  = lane & 15;
  const int m    = m0 + l16;            // row this lane supplies for A
  const int n    = n0 + l16;            // col this lane supplies for B

  const float* af = emb_feat + (size_t)src[m]  * D_SEG + 2 * half;
  const float* ad = emb_dest + (size_t)dest[m] * D_SEG + 2 * half;
  const float* wr = w1 + (size_t)n * D_S + 2 * half;   // w1^T[k][n] = w1[n][k]

  v8f c = {0.f,0.f,0.f,0.f,0.f,0.f,0.f,0.f};

  // K = 0..127 : emb_feat[src] block
#pragma unroll 8
  for (int k0 = 0; k0 < D_SEG; k0 += 4) {
    v2f a = *(const v2f*)(af + k0);
    v2f b = *(const v2f*)(wr + k0);
    c = wmma_f32(a, b, c);
  }
  // K = 128..255 : emb_dest[dest] block
#pragma unroll 8
  for (int k0 = 0; k0 < D_SEG; k0 += 4) {
    v2f a = *(const v2f*)(ad + k0);
    v2f b = *(const v2f*)(wr + D_SEG + k0);
    c = wmma_f32(a, b, c);
  }
  // K = 256..257 : offset[:,1:3]; only half==0 lanes carry valid K, half==1
  // lanes would index K=258,259 (out of bounds) -> supply zeros, no OOB read.
  {
    v2f a = {0.f, 0.f}, b = {0.f, 0.f};
    if (half == 0) {
      a.x = offset[m * 4 + 1];
      a.y = offset[m * 4 + 2];
      b.x = w1[(size_t)n * D_S + 256];
      b.y = w1[(size_t)n * D_S + 257];
    }
    c = wmma_f32(a, b, c);
  }

  // BN(+bias) folded:  relu( c*s + t ),  s = gamma*rsqrt(var+eps),
  //                    t = (b1 - mean)*s + beta
  const float s = bn_gamma[n] * rsqrtf(bn_var[n] + 1e-6f);
  const float t = (b1[n] - bn_mean[n]) * s + bn_beta[n];
#pragma unroll
  for (int vg = 0; vg < 8; ++vg) {
    const int row = m0 + vg + 8 * half;           // C/D layout: M = vg + 8*half
    const float v = fmaf(c[vg], s, t);
    hbuf[(size_t)row * D_M7 + n] = fmaxf(v, 0.f);
  }
}

// ---------------------------------------------------------------------------
// Kernel 2: GEMM2 (h @ w2^T + b2) -> out (4096 x 256)
// One wave32 per 16x16 tile, K = 512.
// ---------------------------------------------------------------------------
__global__ __launch_bounds__(32) void k_gemm2(
    const float* __restrict__ hbuf, const float* __restrict__ w2,
    const float* __restrict__ b2, float* __restrict__ outbuf)
{
  const int n0   = blockIdx.x * 16;
  const int m0   = blockIdx.y * 16;
  const int lane = threadIdx.x & 31;
  const int half = lane >> 4;
  const int l16  = lane & 15;
  const int n    = n0 + l16;

  const float* ar = hbuf + (size_t)(m0 + l16) * D_M7 + 2 * half;
  const float* br = w2   + (size_t)n * D_M7 + 2 * half;   // w2^T[k][n] = w2[n][k]

  v8f c = {0.f,0.f,0.f,0.f,0.f,0.f,0.f,0.f};
#pragma unroll 8
  for (int k0 = 0; k0 < D_M7; k0 += 4) {
    v2f a = *(const v2f*)(ar + k0);
    v2f b = *(const v2f*)(br + k0);
    c = wmma_f32(a, b, c);
  }

  const float bias = b2[n];
#pragma unroll
  for (int vg = 0; vg < 8; ++vg) {
    const int row = m0 + vg + 8 * half;
    outbuf[(size_t)row * D_M8 + n] = c[vg] + bias;
  }
}

// ---------------------------------------------------------------------------
// Kernel 3: per-row gather-dot.  logits[b,m] = dense_w[cand] . out[b,:252]
//           + emb_cand(cos,traffic) . out[b,252:256] + dense_b[cand]
// One block per batch row b; out[b] staged in LDS; 4 lanes per candidate.
// This is the bandwidth-bound stage (~264 MB of dense_w gathers, L2-resident).
// ---------------------------------------------------------------------------
__global__ __launch_bounds__(256) void k_logits(
    const int* __restrict__ dest, const int* __restrict__ t,
    const int* __restrict__ cands,
    const float* __restrict__ dense_w, const float* __restrict__ dense_b,
    const float* __restrict__ cand_w, const float* __restrict__ cand_b,
    const float* __restrict__ segs_src, const float* __restrict__ segs_trg,
    const float* __restrict__ traffic_pop,
    const float* __restrict__ outbuf, float* __restrict__ logits)
{
  __shared__ __align__(16) float s_out[D_M8];
  const int b   = blockIdx.x;
  const int tid = threadIdx.x;

  s_out[tid] = outbuf[(size_t)b * D_M8 + tid];
  __syncthreads();

  const int m    = tid >> 2;   // candidate 0..63
  const int q    = tid & 3;    // quarter of the 252-dot
  const int cand = cands[b * MASK_N + m];

  // dense_w row base = cand*252 floats = cand*1008 B (16B aligned); chunk
  // offsets are 0/256/512/768 B -> float4 loads are aligned.
  const float*  wr  = dense_w + (size_t)cand * DW_COLS;
  const int     c0  = q * 64;
  const int     nf4 = (q == 3) ? 15 : 16;       // last chunk covers 60 cols
  const float4* wp  = (const float4*)(wr + c0);
  const float4* op  = (const float4*)(s_out + c0);

  float acc = 0.f;
  for (int i = 0; i < nf4; ++i) {
    const float4 w = wp[i];
    const float4 o = op[i];
    acc = fmaf(w.x, o.x, acc);
    acc = fmaf(w.y, o.y, acc);
    acc = fmaf(w.z, o.z, acc);
    acc = fmaf(w.w, o.w, acc);
  }
  // reduce the 4 partials within the wave (groups of 4 consecutive lanes)
  acc += __shfl_xor(acc, 1);
  acc += __shfl_xor(acc, 2);

  if (q == 0) {
    const int   db  = dest[b];
    const float pdx = segs_src[(size_t)db * 2];
    const float pdy = segs_src[(size_t)db * 2 + 1];
    const float csx = segs_src[(size_t)cand * 2];
    const float csy = segs_src[(size_t)cand * 2 + 1];
    const float ctx = segs_trg[(size_t)cand * 2];
    const float cty = segs_trg[(size_t)cand * 2 + 1];

    const float v1x = pdx - csx, v1y = pdy - csy;
    const float v2x = ctx - csx, v2y = cty - csy;
    const float num = v1x * v2x + v1y * v2y;
    const float den = fmaxf(sqrtf(v1x * v1x + v1y * v1y) *
                            sqrtf(v2x * v2x + v2y * v2y), 1e-8f);
    const float cosf_   = num / den;
    const float traffic = traffic_pop[(size_t)cand * T_SLOTS + t[b]];

    float tail = 0.f;
#pragma unroll
    for (int j = 0; j < 4; ++j) {
      const float ec = fmaf(cand_w[j * 2], cosf_,
                       fmaf(cand_w[j * 2 + 1], traffic, cand_b[j]));
      tail = fmaf(ec, s_out[DW_COLS + j], tail);
    }
    logits[b * MASK_N + m] = acc + tail + dense_b[cand];
  }
}

// ---------------------------------------------------------------------------
extern "C" void kernel_launch(void* const* d_in, const int* in_sizes, int n_in,
                              void* d_out, int out_size, void* d_ws, size_t ws_size,
                              hipStream_t stream)
{
  const int*   src         = (const int*)  d_in[0];
  const int*   dest        = (const int*)  d_in[1];
  const int*   t           = (const int*)  d_in[2];
  const int*   candidates  = (const int*)  d_in[3];
  const float* offset      = (const float*)d_in[4];
  const float* emb_feat    = (const float*)d_in[5];
  const float* emb_dest    = (const float*)d_in[6];
  const float* w1          = (const float*)d_in[7];
  const float* b1          = (const float*)d_in[8];
  const float* bn_gamma    = (const float*)d_in[9];
  const float* bn_beta     = (const float*)d_in[10];
  const float* bn_mean     = (const float*)d_in[11];
  const float* bn_var      = (const float*)d_in[12];
  const float* w2          = (const float*)d_in[13];
  const float* b2          = (const float*)d_in[14];
  const float* dense_w     = (const float*)d_in[15];
  const float* dense_b     = (const float*)d_in[16];
  const float* cand_w      = (const float*)d_in[17];
  const float* cand_b      = (const float*)d_in[18];
  const float* segs_src    = (const float*)d_in[19];
  const float* segs_trg    = (const float*)d_in[20];
  const float* traffic_pop = (const float*)d_in[21];

  float* logits = (float*)d_out;

  // scratch: h (4096x512 f32, 8 MiB) then out (4096x256 f32, 4 MiB)
  float* hbuf   = (float*)d_ws;
  float* outbuf = hbuf + (size_t)BATCH * D_M7;

  k_gemm1<<<dim3(D_M7 / 16, BATCH / 16), 32, 0, stream>>>(
      src, dest, offset, emb_feat, emb_dest, w1, b1,
      bn_gamma, bn_beta, bn_mean, bn_var, hbuf);

  k_gemm2<<<dim3(D_M8 / 16, BATCH / 16), 32, 0, stream>>>(
      hbuf, w2, b2, outbuf);

  k_logits<<<BATCH, 256, 0, stream>>>(
      dest, t, candidates, dense_w, dense_b, cand_w, cand_b,
      segs_src, segs_trg, traffic_pop, outbuf, logits);
}